// CTCRecognizer_14705968021759
// MI455X (gfx1250) — compile-verified
//
#include <hip/hip_runtime.h>
#include <math.h>

typedef __attribute__((ext_vector_type(2))) float v2f;
typedef __attribute__((ext_vector_type(8))) float v8f;

#define B_   32
#define T_   1024
#define F_   1024
#define V_   56
#define S_   128
#define L_   257
#define NEG_INF_F (-1e30f)

#define TILE_M 128
#define TILE_N 64
#define KT     32            // K-chunk per buffer (double buffered)
#define LDA    36            // padded row stride (floats): conflict-free, 16B-aligned
#define LDW    36
#define NCHUNK (F_ / KT)     // 32
#define A_LOADS (TILE_M * KT / 4)   // 1024 float4 transfers per chunk (4/thread)
#define W_LOADS (TILE_N * KT / 4)   // 512  float4 transfers per chunk (2/thread)

// Async global -> LDS copy of 16 bytes per lane (GLOBAL_LOAD_ASYNC_TO_LDS_B128,
// tracked by ASYNCcnt). lds_off is the LDS byte offset (flat-aperture low 32b).
__device__ __forceinline__ void async_copy_b128(unsigned int lds_off,
                                                const float* gsrc) {
  asm volatile("global_load_async_to_lds_b128 %0, %1, off"
               :: "v"(lds_off), "v"((unsigned long long)(uintptr_t)gsrc)
               : "memory");
}

// ---------------------------------------------------------------------------
// Kernel 1: logits = features x W^T (+bias), then row-wise log-softmax.
// Grid: (B*T)/TILE_M = 256 workgroups, 256 threads (8 waves) each.
// Double-buffered async global->LDS staging overlapped with the f32 WMMA
// K-loop (V_WMMA_F32_16X16X4_F32 keeps full f32 precision, matches reference).
// ---------------------------------------------------------------------------
__global__ __launch_bounds__(256) void gemm_logsoftmax_kernel(
    const float* __restrict__ features,  // [B*T, F]
    const float* __restrict__ W,         // [V, F]
    const float* __restrict__ bias,      // [V]
    float* __restrict__ lp)              // out: [B*T, V] log-probs
{
  __shared__ __align__(16) float Alds[2][TILE_M][LDA];
  __shared__ __align__(16) float Wlds[2][TILE_N][LDW];
  __shared__ float biasS[TILE_N];

  const int tid  = threadIdx.x;
  const int lane = tid & 31;
  const int wave = tid >> 5;           // 0..7
  const int m    = lane & 15;          // M row (A frag) / N col (B frag)
  const int kb   = (lane >> 4) << 1;   // K pair base: 0 (lanes 0-15) or 2
  const size_t rowBase = (size_t)blockIdx.x * TILE_M;

  if (tid < TILE_N) biasS[tid] = (tid < V_) ? bias[tid] : 0.0f;

  // Issue one K-chunk's async loads. Exactly 6 async instructions per wave:
  // uniform trip counts keep the per-wave ASYNCcnt arithmetic exact.
  auto issue_chunk = [&](int k0, int buf) {
    #pragma unroll
    for (int i = tid; i < A_LOADS; i += 256) {       // 4 iterations
      const int r  = i >> 3;
      const int c4 = (i & 7) << 2;
      async_copy_b128((unsigned int)(uintptr_t)&Alds[buf][r][c4],
                      features + (rowBase + (size_t)r) * F_ + k0 + c4);
    }
    #pragma unroll
    for (int i = tid; i < W_LOADS; i += 256) {       // 2 iterations
      const int r  = i >> 3;
      const int c4 = (i & 7) << 2;
      const int sr = (r < V_) ? r : 0;  // padded rows: harmless duplicate of row 0
      async_copy_b128((unsigned int)(uintptr_t)&Wlds[buf][r][c4],
                      W + (size_t)sr * F_ + k0 + c4);
    }
  };

  v8f acc[4] = {};   // 4 N-tiles of 16x16 f32 accumulators

  issue_chunk(0, 0);                       // prologue: chunk 0 -> buf 0

  for (int c = 0; c < NCHUNK; ++c) {
    const int buf = c & 1;
    if (c + 1 < NCHUNK) {
      issue_chunk((c + 1) * KT, buf ^ 1);  // stream next chunk during compute
      // In-order completion: <=6 outstanding means chunk c has fully landed.
      asm volatile("s_wait_asynccnt 0x6" ::: "memory");
    } else {
      asm volatile("s_wait_asynccnt 0x0" ::: "memory");
    }
    __syncthreads();                       // chunk c visible to all waves

    const int rA = wave * 16 + m;
    #pragma unroll
    for (int kk = 0; kk < KT; kk += 4) {
      v2f a;
      a.x = Alds[buf][rA][kk + kb];
      a.y = Alds[buf][rA][kk + kb + 1];
      #pragma unroll
      for (int nt = 0; nt < 4; ++nt) {
        v2f bf;
        bf.x = Wlds[buf][nt * 16 + m][kk + kb];      // B[k][n] = W[n][k]
        bf.y = Wlds[buf][nt * 16 + m][kk + kb + 1];
        acc[nt] = __builtin_amdgcn_wmma_f32_16x16x4_f32(
            /*neg_a=*/false, a, /*neg_b=*/false, bf,
            /*c_mod=*/(short)0, acc[nt],
            /*reuse_a=*/false, /*reuse_b=*/false);
      }
    }
    __syncthreads();   // all waves done with buf before it is refilled
  }

  // Spill D tiles to LDS (reuse A storage; stride 65 -> conflict-free softmax).
  float (*logitsS)[65] = (float (*)[65])&Alds[0][0][0];  // 128*65 <= 2*128*36
  #pragma unroll
  for (int r = 0; r < 8; ++r) {
    const int row = wave * 16 + ((lane < 16) ? r : r + 8);
    logitsS[row][0 * 16 + m] = acc[0][r];
    logitsS[row][1 * 16 + m] = acc[1][r];
    logitsS[row][2 * 16 + m] = acc[2][r];
    logitsS[row][3 * 16 + m] = acc[3][r];
  }
  __syncthreads();

  // Row-wise log-softmax over the 56 valid columns (cols 56..63 never read).
  if (tid < TILE_M) {
    float mx = NEG_INF_F;
    #pragma unroll 8
    for (int n = 0; n < V_; ++n) {
      float v = logitsS[tid][n] + biasS[n];
      mx = fmaxf(mx, v);
    }
    float s = 0.0f;
    #pragma unroll 8
    for (int n = 0; n < V_; ++n) {
      float v = logitsS[tid][n] + biasS[n];
      s += __expf(v - mx);
    }
    const float lse = mx + __logf(s);
    float* dst = lp + (rowBase + (size_t)tid) * V_;
    #pragma unroll 8
    for (int n = 0; n < V_; ++n) {
      dst[n] = (logitsS[tid][n] + biasS[n]) - lse;
    }
  }
}

// ---------------------------------------------------------------------------
// Kernel 2: CTC forward recursion per batch element.
// Grid: B blocks, 288 threads (9 waves) covering L=257 states.
// ---------------------------------------------------------------------------
__global__ __launch_bounds__(288) void ctc_kernel(
    const float* __restrict__ lp,        // [B, T, V] log-probs
    const int* __restrict__ targets,     // [B, S]
    const int* __restrict__ in_len,      // [B]
    const int* __restrict__ tg_len,      // [B]
    float* __restrict__ per_loss)        // [B]
{
  __shared__ int ext[L_];
  __shared__ unsigned char skip[L_];
  __shared__ float alpha[2][L_];

  const int tid = threadIdx.x;
  const int b   = blockIdx.x;
  const int Tin = in_len[b];
  const int tl  = tg_len[b];
  const float* lpb = lp + (size_t)b * T_ * V_;

  for (int l = tid; l < L_; l += blockDim.x)
    ext[l] = (l & 1) ? targets[b * S_ + (l >> 1)] : 0;
  __syncthreads();

  for (int l = tid; l < L_; l += blockDim.x) {
    const int e   = ext[l];
    const int em2 = (l >= 2) ? ext[l - 2] : 0;
    skip[l] = (unsigned char)((e != 0) && (e != em2));
    float a0;
    if (l == 0)       a0 = lpb[e];                           // t = 0, blank
    else if (l == 1)  a0 = (tl > 0) ? lpb[e] : NEG_INF_F;
    else              a0 = NEG_INF_F;
    alpha[0][l] = a0;
  }
  __syncthreads();

  int cur = 0;
  for (int t = 1; t < T_; ++t) {
    const float* lpt = lpb + (size_t)t * V_;
    const int nxt = cur ^ 1;
    if (tid < L_) {
      const float a1 = alpha[cur][tid];
      const float a2 = (tid >= 1) ? alpha[cur][tid - 1] : NEG_INF_F;
      const float a3 = (tid >= 2 && skip[tid]) ? alpha[cur][tid - 2] : NEG_INF_F;
      const float mx = fmaxf(a1, fmaxf(a2, a3));
      const float v  = mx + __logf(__expf(a1 - mx) + __expf(a2 - mx) + __expf(a3 - mx))
                          + lpt[ext[tid]];
      alpha[nxt][tid] = (t < Tin) ? v : a1;
    }
    __syncthreads();
    cur = nxt;
  }

  if (tid == 0) {
    int i1 = 2 * tl - 1; if (i1 < 0) i1 = 0;
    int i2 = 2 * tl;     if (i2 > L_ - 1) i2 = L_ - 1;
    const float l1 = alpha[cur][i1];
    const float l2 = alpha[cur][i2];
    const float mx = fmaxf(l1, l2);
    const float loss = -(mx + __logf(__expf(l1 - mx) + __expf(l2 - mx)));
    const float denom = (tl > 0) ? (float)tl : 1.0f;
    per_loss[b] = loss / denom;
  }
}

// ---------------------------------------------------------------------------
// Kernel 3: mean over batch.
// ---------------------------------------------------------------------------
__global__ void mean_kernel(const float* __restrict__ per_loss,
                            float* __restrict__ out)
{
  if (threadIdx.x == 0) {
    float s = 0.0f;
    for (int b = 0; b < B_; ++b) s += per_loss[b];
    out[0] = s / (float)B_;
  }
}

extern "C" void kernel_launch(void* const* d_in, const int* in_sizes, int n_in,
                              void* d_out, int out_size, void* d_ws, size_t ws_size,
                              hipStream_t stream) {
  (void)in_sizes; (void)n_in; (void)out_size; (void)ws_size;
  const float* features = (const float*)d_in[0];   // [B, T, F]
  const float* W        = (const float*)d_in[1];   // [V, F]
  const float* bias     = (const float*)d_in[2];   // [V]
  const int*   targets  = (const int*)d_in[3];     // [B, S]
  const int*   in_len   = (const int*)d_in[4];     // [B]
  const int*   tg_len   = (const int*)d_in[5];     // [B]

  float* lp       = (float*)d_ws;                  // [B*T, V]
  float* per_loss = lp + (size_t)B_ * T_ * V_;     // [B]

  const int nTiles = (B_ * T_) / TILE_M;  // 256
  gemm_logsoftmax_kernel<<<nTiles, 256, 0, stream>>>(features, W, bias, lp);
  ctc_kernel<<<B_, 288, 0, stream>>>(lp, targets, in_len, tg_len, per_loss);
  mean_kernel<<<1, 32, 0, stream>>>(per_loss, (float*)d_out);
}